// sparse_low_precision_linear_68607807586975
// MI455X (gfx1250) — compile-verified
//
#include <hip/hip_runtime.h>

typedef __attribute__((ext_vector_type(16))) _Float16 v16h;
typedef __attribute__((ext_vector_type(32))) _Float16 v32h;
typedef __attribute__((ext_vector_type(8)))  float    v8f;

#define TOKENS 8192
#define IN_F   4096
#define OUT_F  4096
#define ZP     8
#define TK_TILES (IN_F / 64)    // 64 K-tiles of 64
#define TO_TILES (OUT_F / 16)   // 256 out-row tiles of 16

// ---------------------------------------------------------------------------
// Pack kernel: dequantize + 2:4-compress W^T into SWMMAC-ready layout.
// (one-shot; amortized over the M=8192 token GEMM)
// ---------------------------------------------------------------------------
__global__ __launch_bounds__(256) void pack_w24_68607807586975(
    const int* __restrict__ q, const unsigned char* __restrict__ mask,
    const _Float16* __restrict__ scales,
    _Float16* __restrict__ Apack, unsigned* __restrict__ Aidx)
{
  int tid  = blockIdx.x * 256 + threadIdx.x;
  int lane = tid & 31;
  int tile = tid >> 5;                 // to*TK_TILES + tk
  int to   = tile / TK_TILES;
  int tk   = tile - to * TK_TILES;
  int m    = lane & 15;
  int hs   = lane >> 4;
  int o    = to * 16 + m;
  int kbase = tk * 64;
  float s = (float)scales[o];

  union { v16h v; _Float16 h[16]; } av;

  #pragma unroll
  for (int j = 0; j < 16; ++j) {
    int g  = (j < 8) ? (4 * hs + (j >> 1)) : (8 + 4 * hs + ((j - 8) >> 1));
    int k0 = kbase + 4 * g;
    int i0 = -1, i1 = -1;
    #pragma unroll
    for (int i = 0; i < 4; ++i) {
      if (mask[(size_t)(k0 + i) * OUT_F + o]) {
        if (i0 < 0) i0 = i; else if (i1 < 0) i1 = i;
      }
    }
    if (i0 < 0) i0 = 0;
    if (i1 < 0) i1 = 3;
    int ks = k0 + ((j & 1) ? i1 : i0);
    av.h[j] = (_Float16)((float)(q[(size_t)ks * OUT_F + o] - ZP) * s);
  }

  unsigned idx = 0u;
  #pragma unroll
  for (int gg = 0; gg < 8; ++gg) {
    int g  = 8 * hs + gg;
    int k0 = kbase + 4 * g;
    int i0 = -1, i1 = -1;
    #pragma unroll
    for (int i = 0; i < 4; ++i) {
      if (mask[(size_t)(k0 + i) * OUT_F + o]) {
        if (i0 < 0) i0 = i; else if (i1 < 0) i1 = i;
      }
    }
    if (i0 < 0) i0 = 0;
    if (i1 < 0) i1 = 3;
    idx |= (unsigned)((i0 & 3) | ((i1 & 3) << 2)) << (4 * gg);
  }

  size_t slot = (size_t)tile * 32 + lane;
  *(((v16h*)Apack) + slot) = av.v;
  Aidx[slot] = idx;
}

// ---------------------------------------------------------------------------
// Async staging helpers (CDNA5 GLOBAL_LOAD_ASYNC_TO_LDS, tracked by ASYNCcnt).
// INST_OFFSET is added to both the global and LDS addresses, so a 32B copy is
// two b128 ops sharing the same address VGPRs.
// ---------------------------------------------------------------------------
__device__ __forceinline__ void async_stage32B(unsigned lds_off, const _Float16* g) {
  unsigned long long ga = (unsigned long long)(size_t)g;
  asm volatile(
      "global_load_async_to_lds_b128 %0, %1, off\n\t"
      "global_load_async_to_lds_b128 %0, %1, off offset:16"
      :: "v"(lds_off), "v"(ga) : "memory");
}
__device__ __forceinline__ void wait_async0() {
  asm volatile("s_wait_asynccnt 0x0" ::: "memory");
}
__device__ __forceinline__ void sched_fence() {
#if __has_builtin(__builtin_amdgcn_sched_barrier)
  __builtin_amdgcn_sched_barrier(0);
#endif
}

// ---------------------------------------------------------------------------
// SWMMAC GEMM: y^T = A(sparse W^T) x B(x^T), double-buffered async x staging.
// ---------------------------------------------------------------------------
__global__ __launch_bounds__(256) void swmmac_gemm_68607807586975(
    const _Float16* __restrict__ x,
    const _Float16* __restrict__ Apack,
    const unsigned* __restrict__ Aidx,
    _Float16* __restrict__ y)
{
  __shared__ _Float16 xs[2][64][72];   // double buffer; 72-half padded rows

  int lane = threadIdx.x & 31;
  int wave = threadIdx.x >> 5;
  int m    = lane & 15;
  int hs   = lane >> 4;
  int obase = blockIdx.x * 128 + wave * 16;
  int tbase = blockIdx.y * 64;
  int to    = obase >> 4;

  const v16h* Ap = (const v16h*)Apack;

  v8f acc[4] = {};

  // staging assignment: thread copies 32B (16 halves) of one token row
  int tkn = threadIdx.x >> 2;
  int kc  = (threadIdx.x & 3) << 4;
  const _Float16* xrow = x + (size_t)(tbase + tkn) * IN_F + kc;
  // low 32 bits of a flat LDS address are the LDS byte offset
  unsigned loff0 = (unsigned)(size_t)(void*)&xs[0][tkn][kc];
  unsigned loff1 = (unsigned)(size_t)(void*)&xs[1][tkn][kc];

  union Bu { v32h v; float4 f[4]; };

#define COMPUTE_TILE(BUF, TK)                                                  \
  {                                                                            \
    size_t slot = ((size_t)to * TK_TILES + (TK)) * 32 + lane;                  \
    v16h a  = Ap[slot];                                                        \
    int  sx = (int)Aidx[slot];                                                 \
    __builtin_prefetch(Ap + slot + 64, 0, 3);                                  \
    Bu b0, b1, b2, b3;                                                         \
    const float4* p0 = (const float4*)&xs[BUF][0 * 16 + m][hs * 16];           \
    const float4* p1 = (const float4*)&xs[BUF][1 * 16 + m][hs * 16];           \
    const float4* p2 = (const float4*)&xs[BUF][2 * 16 + m][hs * 16];           \
    const float4* p3 = (const float4*)&xs[BUF][3 * 16 + m][hs * 16];           \
    b0.f[0] = p0[0]; b0.f[1] = p0[1]; b0.f[2] = p0[4]; b0.f[3] = p0[5];        \
    b1.f[0] = p1[0]; b1.f[1] = p1[1]; b1.f[2] = p1[4]; b1.f[3] = p1[5];        \
    b2.f[0] = p2[0]; b2.f[1] = p2[1]; b2.f[2] = p2[4]; b2.f[3] = p2[5];        \
    b3.f[0] = p3[0]; b3.f[1] = p3[1]; b3.f[2] = p3[4]; b3.f[3] = p3[5];        \
    sched_fence();   /* keep all 16 ds_loads ahead of the swmmac group */      \
    acc[0] = __builtin_amdgcn_swmmac_f32_16x16x64_f16(                         \
                 false, a, false, b0.v, acc[0], sx, false, false);             \
    acc[1] = __builtin_amdgcn_swmmac_f32_16x16x64_f16(                         \
                 false, a, false, b1.v, acc[1], sx, false, false);             \
    acc[2] = __builtin_amdgcn_swmmac_f32_16x16x64_f16(                         \
                 false, a, false, b2.v, acc[2], sx, false, false);             \
    acc[3] = __builtin_amdgcn_swmmac_f32_16x16x64_f16(                         \
                 false, a, false, b3.v, acc[3], sx, false, false);             \
  }

  // prologue: DMA tile 0 into buffer 0
  async_stage32B(loff0, xrow);

  for (int tk = 0; tk < TK_TILES; tk += 2) {
    // ---- even tile: compute buf0, DMA tile tk+1 into buf1 ----
    wait_async0();        // my tile-tk loads landed (in-order completion)
    __syncthreads();      // everyone's landed; everyone done reading buf1
    async_stage32B(loff1, xrow + (size_t)(tk + 1) * 64);
    COMPUTE_TILE(0, tk)

    // ---- odd tile: compute buf1, DMA tile tk+2 into buf0 ----
    wait_async0();
    __syncthreads();
    if (tk + 2 < TK_TILES)
      async_stage32B(loff0, xrow + (size_t)(tk + 2) * 64);
    COMPUTE_TILE(1, tk + 1)
  }
#undef COMPUTE_TILE

  // D tile: VGPR r, lanes 0-15 -> M=r, lanes 16-31 -> M=r+8; N = token = lane&15.
  // y is [TOKENS, OUT_F]; the 8 M-values per lane are contiguous in o -> b128.
  size_t outb = (size_t)tbase * OUT_F + (size_t)obase + 8 * hs;
  #pragma unroll
  for (int tt = 0; tt < 4; ++tt) {
    union { float4 f; _Float16 h[8]; } ov;
    #pragma unroll
    for (int r = 0; r < 8; ++r) ov.h[r] = (_Float16)acc[tt][r];
    *(float4*)(y + outb + (size_t)(tt * 16 + m) * OUT_F) = ov.f;
  }
}

extern "C" void kernel_launch(void* const* d_in, const int* in_sizes, int n_in,
                              void* d_out, int out_size, void* d_ws, size_t ws_size,
                              hipStream_t stream) {
  const _Float16*      xp = (const _Float16*)d_in[0];
  const int*           qp = (const int*)d_in[1];
  const unsigned char* mp = (const unsigned char*)d_in[2];
  const _Float16*      sp = (const _Float16*)d_in[3];
  _Float16*            yp = (_Float16*)d_out;

  // workspace: 16 MB packed A (f16) + 2 MB indices (u32)
  _Float16* Apack = (_Float16*)d_ws;
  unsigned* Aidx  = (unsigned*)((char*)d_ws + (size_t)OUT_F * (IN_F / 2) * 2u);

  int packThreads = TO_TILES * TK_TILES * 32;   // 524288
  pack_w24_68607807586975<<<packThreads / 256, 256, 0, stream>>>(
      qp, mp, sp, Apack, Aidx);

  dim3 grid(OUT_F / 128, TOKENS / 64);
  swmmac_gemm_68607807586975<<<grid, 256, 0, stream>>>(xp, Apack, Aidx, yp);
}